// ToiPoolingConv_23390391894068
// MI455X (gfx1250) — compile-verified
//
#include <hip/hip_runtime.h>
#include <cstdint>

// ---------------------------------------------------------------------------
// ToI pooling (ROI bilinear pool) for MI455X / gfx1250.
// Shapes fixed by the reference setup:
//   img:  (B=2, T=16, H=64, W=64, C=256) fp32
//   rois: (T=16, R=32, 4) int32   [x0, y0, w, h]
//   out:  (B, R, T, P=7, P=7, C) fp32
// Memory-bound gather: B128 coalesced loads along C, NT B128 stores,
// gfx1250 async-to-LDS staging of ROI descriptors, global_prefetch_b8.
// ---------------------------------------------------------------------------

typedef float v4f __attribute__((ext_vector_type(4)));
typedef int   v4i __attribute__((vector_size(16)));   // matches builtin proto

#define DIM_B 2
#define DIM_T 16
#define DIM_H 64
#define DIM_W 64
#define DIM_C 256
#define DIM_R 32
#define DIM_P 7

#define NCELLS (DIM_B * DIM_R * DIM_T * DIM_P * DIM_P)  // 50176
#define CELLS_PER_BLOCK 4
#define NBLOCKS (NCELLS / CELLS_PER_BLOCK)              // 12544 (exact)

#if defined(__gfx1250__) &&                                                   \
    __has_builtin(__builtin_amdgcn_global_load_async_to_lds_b128) &&          \
    __has_builtin(__builtin_amdgcn_s_wait_asynccnt)
#define USE_ASYNC_ROI 1
#else
#define USE_ASYNC_ROI 0
#endif

#if USE_ASYNC_ROI
typedef __attribute__((address_space(1))) v4i* gbl_v4i_p;  // global (AS1)
typedef __attribute__((address_space(3))) v4i* lds_v4i_p;  // LDS (AS3)
#endif

__global__ __launch_bounds__(256) void toi_pool_kernel(
    const float* __restrict__ img,
    const int*   __restrict__ rois,
    float*       __restrict__ out)
{
    const int tid  = threadIdx.x;
    const int grp  = tid >> 6;   // 0..3 : pooled cell within block (2 waves each)
    const int lane = tid & 63;   // 0..63: 4 channels per lane (float4 over C=256)

    const int cell = blockIdx.x * CELLS_PER_BLOCK + grp;   // < NCELLS always

    // decode (b, r, t, py, px) — row-major over (B, R, T, P, P)
    int tmp = cell;
    const int px = tmp % DIM_P; tmp /= DIM_P;
    const int py = tmp % DIM_P; tmp /= DIM_P;
    const int t  = tmp % DIM_T; tmp /= DIM_T;
    const int r  = tmp % DIM_R; tmp /= DIM_R;
    const int b  = tmp;

    // -------- ROI descriptor: rois[t][r][0..3] = x0, y0, w, h --------------
    const int roi_off = 4 * (t * DIM_R + r);
    int x0, y0, ww, hh;
#if USE_ASYNC_ROI
    __shared__ int sroi[CELLS_PER_BLOCK][4];
    if (lane == 0) {
        // gfx1250 ASYNC path: 16B ROI descriptor DMA'd straight into LDS.
        __builtin_amdgcn_global_load_async_to_lds_b128(
            (gbl_v4i_p)(uintptr_t)(rois + roi_off),
            (lds_v4i_p)(uintptr_t)(&sroi[grp][0]),
            /*offset=*/0, /*cpol=*/0);
        __builtin_amdgcn_s_wait_asynccnt(0);
    }
    __syncthreads();
    x0 = sroi[grp][0]; y0 = sroi[grp][1]; ww = sroi[grp][2]; hh = sroi[grp][3];
#else
    const int4 roi = *reinterpret_cast<const int4*>(rois + roi_off);
    x0 = roi.x; y0 = roi.y; ww = roi.z; hh = roi.w;
#endif

    // -------- bilinear sample coordinates (matches reference math) ---------
    const float sy  = (float)hh / (float)DIM_P;
    const float srcy = (float)py * sy;
    const int   ylo_i = (int)srcy;                 // srcy >= 0 -> trunc == floor
    const float wy  = srcy - (float)ylo_i;
    const int   yl  = y0 + ylo_i;
    const int   yh  = y0 + ((ylo_i + 1 < hh) ? (ylo_i + 1) : (hh - 1));

    const float sx  = (float)ww / (float)DIM_P;
    const float srcx = (float)px * sx;
    const int   xlo_i = (int)srcx;
    const float wx  = srcx - (float)xlo_i;
    const int   xl  = x0 + xlo_i;
    const int   xh  = x0 + ((xlo_i + 1 < ww) ? (xlo_i + 1) : (ww - 1));

    // -------- gather 4 corner channel-vectors (coalesced B128 along C) -----
    const size_t base = ((size_t)b * DIM_T + (size_t)t) * (size_t)(DIM_H * DIM_W * DIM_C);
    const size_t o00 = base + (size_t)(yl * DIM_W + xl) * DIM_C;
    const size_t o01 = base + (size_t)(yl * DIM_W + xh) * DIM_C;
    const size_t o10 = base + (size_t)(yh * DIM_W + xl) * DIM_C;
    const size_t o11 = base + (size_t)(yh * DIM_W + xh) * DIM_C;

    // gfx1250 prefetch of the far corner row while the near loads issue.
    if (lane == 0) {
        __builtin_prefetch(img + o11, 0, 0);   // -> global_prefetch_b8
    }

    const int c4 = lane << 2;  // channel offset (4 floats per lane)
    const v4f v00 = *reinterpret_cast<const v4f*>(img + o00 + c4);
    const v4f v01 = *reinterpret_cast<const v4f*>(img + o01 + c4);
    const v4f v10 = *reinterpret_cast<const v4f*>(img + o10 + c4);
    const v4f v11 = *reinterpret_cast<const v4f*>(img + o11 + c4);

    // -------- bilinear blend (same operation order as the reference) -------
    const v4f top = v00 + (v01 - v00) * wx;
    const v4f bot = v10 + (v11 - v10) * wx;
    const v4f res = top + (bot - top) * wy;

    // -------- streaming store: keep output out of L2 (img stays resident) --
    float* dst = out + (size_t)cell * DIM_C + c4;
    __builtin_nontemporal_store(res, reinterpret_cast<v4f*>(dst));
}

extern "C" void kernel_launch(void* const* d_in, const int* in_sizes, int n_in,
                              void* d_out, int out_size, void* d_ws, size_t ws_size,
                              hipStream_t stream) {
    (void)in_sizes; (void)n_in; (void)out_size; (void)d_ws; (void)ws_size;
    const float* img  = (const float*)d_in[0];
    const int*   rois = (const int*)d_in[1];
    // d_in[2] = pool_size (==7), baked into the kernel as DIM_P.
    float* out = (float*)d_out;

    toi_pool_kernel<<<dim3(NBLOCKS), dim3(256), 0, stream>>>(img, rois, out);
}